// Group_38216619000510
// MI455X (gfx1250) — compile-verified
//
#include <hip/hip_runtime.h>
#include <hip/hip_bf16.h>

#define BATCH 16
#define NPTS  8192
#define NGRP  256
#define KNN   32
#define CH    512   // points per LDS chunk in the WMMA top-k kernel

typedef float v2f __attribute__((ext_vector_type(2)));
typedef float v8f __attribute__((ext_vector_type(8)));

// ---------------------------------------------------------------------------
// K0: per-point squared norms ||p||^2  (feeds the C matrix of the WMMA)
// ---------------------------------------------------------------------------
__global__ void ptsq_kernel(const float* __restrict__ xyz, float* __restrict__ ptsq) {
    int i = blockIdx.x * 256 + threadIdx.x;
    if (i < BATCH * NPTS) {
        const float* P = xyz + (size_t)i * 3;
        __builtin_prefetch(P + 256 * 3, 0, 1);   // global_prefetch_b8
        ptsq[i] = P[0] * P[0] + P[1] * P[1] + P[2] * P[2];
    }
}

// ---------------------------------------------------------------------------
// K1: farthest point sampling. One block per batch, 1024 threads, 8 points
// per thread held in registers; 255 serial steps of min-update + LDS argmax.
// Emits centersA = (-2x,-2y,-2z,||c||^2) (WMMA A rows) and centersP=(x,y,z,0).
// ---------------------------------------------------------------------------
__global__ __launch_bounds__(1024) void fps_kernel(const float* __restrict__ xyz,
                                                   float* __restrict__ cA,
                                                   float* __restrict__ cP) {
    int b = blockIdx.x;
    int tid = threadIdx.x;
    const float* X = xyz + (size_t)b * NPTS * 3;

    float px[8], py[8], pz[8], dmin[8];
#pragma unroll
    for (int j = 0; j < 8; ++j) {
        int i = tid * 8 + j;
        px[j] = X[i * 3 + 0];
        py[j] = X[i * 3 + 1];
        pz[j] = X[i * 3 + 2];
        dmin[j] = __builtin_inff();
    }

    __shared__ float lp[3];
    __shared__ float sval[1024];
    __shared__ int   sidx[1024];

    if (tid == 0) {
        float x = X[0], y = X[1], z = X[2];
        size_t o = ((size_t)b * NGRP + 0) * 4;
        cA[o + 0] = -2.0f * x; cA[o + 1] = -2.0f * y; cA[o + 2] = -2.0f * z;
        cA[o + 3] = x * x + y * y + z * z;
        cP[o + 0] = x; cP[o + 1] = y; cP[o + 2] = z; cP[o + 3] = 0.0f;
        lp[0] = x; lp[1] = y; lp[2] = z;
    }
    __syncthreads();

    for (int g = 1; g < NGRP; ++g) {
        float lx = lp[0], ly = lp[1], lz = lp[2];
        float bv = -1.0f; int bi = tid * 8;
#pragma unroll
        for (int j = 0; j < 8; ++j) {
            float dx = px[j] - lx, dy = py[j] - ly, dz = pz[j] - lz;
            float d = dx * dx + dy * dy + dz * dz;
            dmin[j] = fminf(dmin[j], d);
            if (dmin[j] > bv) { bv = dmin[j]; bi = tid * 8 + j; }
        }
        sval[tid] = bv; sidx[tid] = bi;
        __syncthreads();
        for (int s = 512; s > 0; s >>= 1) {
            if (tid < s) {
                float ov = sval[tid + s]; int oi = sidx[tid + s];
                if (ov > sval[tid] || (ov == sval[tid] && oi < sidx[tid])) {
                    sval[tid] = ov; sidx[tid] = oi;
                }
            }
            __syncthreads();
        }
        if (tid == 0) {
            int pi = sidx[0];
            float x = X[pi * 3], y = X[pi * 3 + 1], z = X[pi * 3 + 2];
            size_t o = ((size_t)b * NGRP + g) * 4;
            cA[o + 0] = -2.0f * x; cA[o + 1] = -2.0f * y; cA[o + 2] = -2.0f * z;
            cA[o + 3] = x * x + y * y + z * z;
            cP[o + 0] = x; cP[o + 1] = y; cP[o + 2] = z; cP[o + 3] = 0.0f;
            lp[0] = x; lp[1] = y; lp[2] = z;
        }
        __syncthreads();
    }
}

// ---------------------------------------------------------------------------
// K2: distance matrix via V_WMMA_F32_16X16X4_F32 + exact online top-32.
// Block = (batch, group-of-16-centers): 16 waves. Each wave computes two
// 16x16 d^2 tiles per 512-point chunk (D = A*B + C gives d^2 directly),
// spills to LDS, then wave w owns center row w and merges the chunk into a
// per-lane 32-entry best list (K == wave32) with early-exit extraction.
// ---------------------------------------------------------------------------
__global__ __launch_bounds__(512) void topk_kernel(const float* __restrict__ xyz,
                                                   const float* __restrict__ ptsq,
                                                   const float* __restrict__ cA,
                                                   const float* __restrict__ cP,
                                                   float* __restrict__ tmp_nb) {
    int b  = blockIdx.x >> 4;
    int cg = blockIdx.x & 15;
    int m0 = cg * 16;
    int tid  = threadIdx.x;
    int lane = tid & 31;
    int w    = tid >> 5;

    __shared__ float s_d2[16 * CH];

    const float* Xb = xyz  + (size_t)b * NPTS * 3;
    const float* Qb = ptsq + (size_t)b * NPTS;

    // A fragment: 16x4 f32 -> 2 VGPRs. lanes 0-15: K=0,1; lanes 16-31: K=2,3.
    int mr = lane & 15;
    const float* Ar = cA + ((size_t)b * NGRP + m0 + mr) * 4;
    int koff = (lane >> 4) * 2;
    v2f afrag;
    afrag.x = Ar[koff + 0];
    afrag.y = Ar[koff + 1];

    float bestd = __builtin_inff();
    int   besti = 0;

    for (int ch = 0; ch < NPTS; ch += CH) {
        // ---- build B/C fragments and run WMMA (2 tiles per wave) ----
#pragma unroll
        for (int tt = 0; tt < 2; ++tt) {
            int colbase = w * 32 + tt * 16;
            int p = ch + colbase + (lane & 15);
            const float* P = Xb + (size_t)p * 3;
            float pxv = P[0], pyv = P[1], pzv = P[2];
            bool hi = lane >= 16;
            v2f bfrag;                    // B col (px,py,pz,1): lanes0-15 K=0,1; lanes16-31 K=2,3
            bfrag.x = hi ? pzv : pxv;
            bfrag.y = hi ? 1.0f : pyv;
            float pq = Qb[p];
            v8f cacc;
#pragma unroll
            for (int v = 0; v < 8; ++v) cacc[v] = pq;   // C[m][n] = ||p_n||^2

            v8f dres = __builtin_amdgcn_wmma_f32_16x16x4_f32(
                false, afrag, false, bfrag, (short)0, cacc, false, false);

            int mbase = hi ? 8 : 0;
            int col = colbase + (lane & 15);
#pragma unroll
            for (int v = 0; v < 8; ++v)
                s_d2[(mbase + v) * CH + col] = dres[v];
        }
        if (ch + CH < NPTS)
            __builtin_prefetch(Xb + (size_t)(ch + CH) * 3 + lane * 12, 0, 1);
        __syncthreads();

        // ---- merge chunk into running top-32 (wave w owns center row w) ----
        float* row = s_d2 + w * CH;
        for (int k = 0; k < KNN; ++k) {
            // wave-parallel argmin over the chunk
            float md = __builtin_inff(); int mi = 0;
            for (int j = lane; j < CH; j += 32) {
                float dv = row[j];
                if (dv < md) { md = dv; mi = j; }
            }
#pragma unroll
            for (int off = 16; off; off >>= 1) {
                float od = __shfl_xor(md, off, 32);
                int   oi = __shfl_xor(mi, off, 32);
                if (od < md || (od == md && oi < mi)) { md = od; mi = oi; }
            }
            // current worst of the 32-entry best list (one entry per lane)
            float xm = bestd; int xl = lane;
#pragma unroll
            for (int off = 16; off; off >>= 1) {
                float om = __shfl_xor(xm, off, 32);
                int   ol = __shfl_xor(xl, off, 32);
                if (om > xm || (om == xm && ol < xl)) { xm = om; xl = ol; }
            }
            if (md >= xm) break;            // chunk is ascending: nothing left can improve
            if (lane == xl) { bestd = md; besti = ch + mi; }
            if (lane == 0) row[mi] = __builtin_inff();
        }
        __syncthreads();
    }

    // ---- sort the 32 winners ascending and emit recentred neighbors ----
    const float* Cc = cP + ((size_t)b * NGRP + m0 + w) * 4;
    float cx = Cc[0], cy = Cc[1], cz = Cc[2];
    float* Ob = tmp_nb + (size_t)(b * NGRP + m0 + w) * KNN * 3;

    float myd = bestd; int myi = besti;
    for (int k = 0; k < KNN; ++k) {
        float md = myd; int ml = lane;
#pragma unroll
        for (int off = 16; off; off >>= 1) {
            float od = __shfl_xor(md, off, 32);
            int   ol = __shfl_xor(ml, off, 32);
            if (od < md || (od == md && ol < ml)) { md = od; ml = ol; }
        }
        if (lane == ml) {
            const float* P = Xb + (size_t)myi * 3;
            Ob[k * 3 + 0] = P[0] - cx;
            Ob[k * 3 + 1] = P[1] - cy;
            Ob[k * 3 + 2] = P[2] - cz;
            myd = __builtin_inff();
        }
    }
}

// ---------------------------------------------------------------------------
// K3: greedy nearest-neighbor ordering over the 256 centers. One block per
// batch, one center per thread (visited flag in a register), 255 serial
// argmin steps through LDS.
// ---------------------------------------------------------------------------
__global__ __launch_bounds__(256) void order_kernel(const float* __restrict__ cP,
                                                    int* __restrict__ order) {
    int b = blockIdx.x;
    int g = threadIdx.x;
    size_t o = ((size_t)b * NGRP + g) * 4;
    float cx = cP[o + 0], cy = cP[o + 1], cz = cP[o + 2];
    bool visited = (g == 0);

    __shared__ float lp[3];
    __shared__ float sval[NGRP];
    __shared__ int   sidx[NGRP];

    if (g == 0) {
        order[b * NGRP] = 0;
        lp[0] = cx; lp[1] = cy; lp[2] = cz;
    }
    __syncthreads();

    for (int i = 1; i < NGRP; ++i) {
        float lx = lp[0], ly = lp[1], lz = lp[2];
        float dx = cx - lx, dy = cy - ly, dz = cz - lz;
        float d = visited ? __builtin_inff() : (dx * dx + dy * dy + dz * dz);
        sval[g] = d; sidx[g] = g;
        __syncthreads();
        for (int s = 128; s > 0; s >>= 1) {
            if (g < s) {
                float ov = sval[g + s]; int oi = sidx[g + s];
                if (ov < sval[g] || (ov == sval[g] && oi < sidx[g])) {
                    sval[g] = ov; sidx[g] = oi;
                }
            }
            __syncthreads();
        }
        int nxt = sidx[0];
        if (g == nxt) {
            visited = true;
            order[b * NGRP + i] = g;
            lp[0] = cx; lp[1] = cy; lp[2] = cz;
        }
        __syncthreads();
    }
}

// ---------------------------------------------------------------------------
// K4: apply the greedy order: out = (nb[order], center[order]) flat-concat.
// ---------------------------------------------------------------------------
__global__ __launch_bounds__(128) void gather_kernel(const float* __restrict__ tmp_nb,
                                                     const float* __restrict__ cP,
                                                     const int* __restrict__ order,
                                                     float* __restrict__ out) {
    int bg = blockIdx.x;            // b*NGRP + g
    int b = bg >> 8;
    int gs = order[bg];             // source group
    int t = threadIdx.x;
    size_t src = (size_t)(b * NGRP + gs) * (KNN * 3);
    size_t dst = (size_t)bg * (KNN * 3);
    if (t < KNN * 3) {
        out[dst + t] = tmp_nb[src + t];
    } else if (t < KNN * 3 + 3) {
        int c = t - KNN * 3;
        out[(size_t)BATCH * NGRP * KNN * 3 + (size_t)bg * 3 + c] =
            cP[(size_t)(b * NGRP + gs) * 4 + c];
    }
}

// ---------------------------------------------------------------------------
// Launch
// ---------------------------------------------------------------------------
extern "C" void kernel_launch(void* const* d_in, const int* in_sizes, int n_in,
                              void* d_out, int out_size, void* d_ws, size_t ws_size,
                              hipStream_t stream) {
    const float* xyz = (const float*)d_in[0];
    float* out = (float*)d_out;

    // workspace layout (floats): cA | cP | ptsq | tmp_nb | order(int)
    float* cA     = (float*)d_ws;                       // B*G*4   = 16384
    float* cP     = cA + (size_t)BATCH * NGRP * 4;      // B*G*4   = 16384
    float* ptsq   = cP + (size_t)BATCH * NGRP * 4;      // B*N     = 131072
    float* tmp_nb = ptsq + (size_t)BATCH * NPTS;        // B*G*K*3 = 393216
    int*   order  = (int*)(tmp_nb + (size_t)BATCH * NGRP * KNN * 3);

    ptsq_kernel<<<(BATCH * NPTS + 255) / 256, 256, 0, stream>>>(xyz, ptsq);
    fps_kernel<<<BATCH, 1024, 0, stream>>>(xyz, cA, cP);
    topk_kernel<<<BATCH * (NGRP / 16), 512, 0, stream>>>(xyz, ptsq, cA, cP, tmp_nb);
    order_kernel<<<BATCH, NGRP, 0, stream>>>(cP, order);
    gather_kernel<<<BATCH * NGRP, 128, 0, stream>>>(tmp_nb, cP, order, out);
}